// F81SupervisedLoss_21861383536666
// MI455X (gfx1250) — compile-verified
//
#include <hip/hip_runtime.h>
#include <stdint.h>

// F81 supervised loss: Felsenstein pruning on a balanced 64-leaf tree, two
// equilibrium distributions (softmax(logits), normalized pi_true), masked mean
// of ll_true - ll_pred.
//
// MI455X / gfx1250 strategy:
//  - one site per lane (wave32); depth-first pruning with 7-entry register stack
//  - BOTH trees packed into float2 {pred, true}; packed math expressed as
//    native <2 x float> IR (llvm.fma.v2f32 via __builtin_elementwise_fma) so
//    ISel can select v_pk_{fma,mul,add}_f32 with full RA/scheduler awareness
//    (inline-asm pk ops in rounds 3-6 caused allocator spills)
//  - msa_codes staged per-wave into LDS via global_load_async_to_lds_b128
//    (coalesced 512B/instruction), padded rows (stride 272) vs bank conflicts
//  - branch coefficients exp(-t) precomputed once; uniform LDS broadcasts
//  - deterministic reduction: wave shfl_xor -> block partial -> finalize kernel

typedef float v2f __attribute__((ext_vector_type(2)));   // {pred, true}

#define N_STATES   4
#define N_LEAVES   64
#define N_SITES    (16 * 8192)          // B * L = 131072
#define EPS_F      1e-12f

#define BLOCK_T    128
#define WAVES_PB   4
#define SITES_PW   32
#define SITES_PB   (WAVES_PB * SITES_PW)   // 128
#define N_BLOCKS   (N_SITES / SITES_PB)    // 1024
#define ROW_DW     68                      // 272B per site row (64 codes + pad)
#define SITE_STRIDE (ROW_DW * 4)           // 272 bytes, 16B aligned
#define TILE_BYTES (SITES_PW * SITE_STRIDE) // 8704 per wave

// ws layout (floats): [0..125] A=exp(-t); [128..128+NB) partial sums;
// [128+NB..128+2NB) partial counts.
#define WS_A     0
#define WS_SUM   128
#define WS_CNT   (128 + N_BLOCKS)

// ---- packed helpers: pure vector IR, no inline asm ----
__device__ __forceinline__ v2f pk_mul(v2f a, v2f b) { return a * b; }
__device__ __forceinline__ v2f pk_add(v2f a, v2f b) { return a + b; }
__device__ __forceinline__ v2f pk_fma(v2f a, v2f b, v2f c) {
  return __builtin_elementwise_fma(a, b, c);      // llvm.fma.v2f32
}
__device__ __forceinline__ v2f splat2(float x) { return (v2f){x, x}; }
__device__ __forceinline__ v2f vmax2(v2f a, float b) {
  v2f r; r.x = fmaxf(a.x, b); r.y = fmaxf(a.y, b); return r;
}
__device__ __forceinline__ v2f vlog2c(v2f a) {
  v2f r; r.x = __logf(a.x); r.y = __logf(a.y); return r;
}
__device__ __forceinline__ v2f vrcp2(v2f a) {
  v2f r; r.x = __builtin_amdgcn_rcpf(a.x); r.y = __builtin_amdgcn_rcpf(a.y);
  return r;
}

__global__ __launch_bounds__(BLOCK_T, 1) void f81_main_kernel(
    const float* __restrict__ logits, const int* __restrict__ codes,
    const float* __restrict__ pit_g, const unsigned char* __restrict__ vmask,
    float* __restrict__ ws) {
  __shared__ uint32_t tile[WAVES_PB * SITES_PW * ROW_DW];
  __shared__ float Ash[2 * N_LEAVES - 2];
  __shared__ float wred[2 * WAVES_PB];

  const int tid  = threadIdx.x;
  const int lane = tid & 31;
  const int wv   = tid >> 5;
  const int waveSite0 = blockIdx.x * SITES_PB + wv * SITES_PW;
  const int site = waveSite0 + lane;

  if (tid < 2 * N_LEAVES - 2) Ash[tid] = ws[WS_A + tid];

  // ---- async stage: this wave's 32 sites x 64 int32 codes -> LDS ----
  {
    const int* gbase = codes + (size_t)waveSite0 * N_LEAVES;
    uint32_t goff  = (uint32_t)(lane * 16);
    uint32_t laddr = (uint32_t)(uintptr_t)&tile[0]
                   + (uint32_t)(wv * TILE_BYTES)
                   + (uint32_t)((lane >> 4) * SITE_STRIDE + (lane & 15) * 16);
#pragma unroll
    for (int k = 0; k < 16; ++k) {
      asm volatile("global_load_async_to_lds_b128 %0, %1, %2 offset:0"
                   :: "v"(laddr), "v"(goff), "s"(gbase)
                   : "memory");
      goff  += 512u;
      laddr += 2u * SITE_STRIDE;
    }
  }

  __syncthreads();   // Ash ready (each wave consumes only its own code tile)

  // Overlap with async copies: softmax(logits) and pi_true normalization.
  float4 lg = ((const float4*)logits)[site];
  float mx = fmaxf(fmaxf(lg.x, lg.y), fmaxf(lg.z, lg.w));
  float e0 = __expf(lg.x - mx), e1 = __expf(lg.y - mx),
        e2 = __expf(lg.z - mx), e3 = __expf(lg.w - mx);
  float rs = __builtin_amdgcn_rcpf(e0 + e1 + e2 + e3);

  float4 pvv = ((const float4*)pit_g)[site];
  float q0 = fmaxf(pvv.x, EPS_F), q1 = fmaxf(pvv.y, EPS_F),
        q2 = fmaxf(pvv.z, EPS_F), q3 = fmaxf(pvv.w, EPS_F);
  float rq = __builtin_amdgcn_rcpf(q0 + q1 + q2 + q3);

  // pi[j] = { softmax_pred_j , normalized_true_j }
  v2f pi[4];
  pi[0] = (v2f){e0 * rs, q0 * rq};
  pi[1] = (v2f){e1 * rs, q1 * rq};
  pi[2] = (v2f){e2 * rs, q2 * rq};
  pi[3] = (v2f){e3 * rs, q3 * rq};

  asm volatile("s_wait_asynccnt 0" ::: "memory");

  // typed shared pointer -> guaranteed ds_load_b128 reads
  const uint4* row = (const uint4*)&tile[(wv * SITES_PW + lane) * ROW_DW];

  // Depth-first Felsenstein pruning; both trees ride the two f32 halves.
  v2f stk[7][4];
  v2f ls = (v2f){0.f, 0.f};
  v2f ll = (v2f){0.f, 0.f};
  const v2f one2  = (v2f){1.f, 1.f};
  const v2f zero2 = (v2f){0.f, 0.f};
  int depth = 0;
  const int LOFF[6] = {0, 64, 96, 112, 120, 124};  // per-level coef offsets

#pragma unroll
  for (int s4 = 0; s4 < 16; ++s4) {
    uint4 qv = row[s4];
    unsigned cs[4] = {qv.x, qv.y, qv.z, qv.w};
#pragma unroll
    for (int u = 0; u < 4; ++u) {
      const int s = s4 * 4 + u;
      const unsigned c = cs[u];
      const bool gap = (c >= 4u);
      float a  = Ash[s];
      v2f a2  = splat2(a);
      v2f oa2 = splat2(1.0f - a);
      // leaf message: gap -> ones; else a*onehot + (1-a)*pi   (pk fma)
#pragma unroll
      for (int j = 0; j < 4; ++j) {
        v2f addv = (c == (unsigned)j) ? a2 : zero2;
        v2f m = pk_fma(oa2, pi[j], addv);
        stk[depth][j] = gap ? one2 : m;
      }
      ++depth;
      // merge while trailing bits of (s+1) allow (fully constant-folded)
#pragma unroll
      for (int d = 1; d <= 6; ++d) {
        if (((s + 1) & ((1 << d) - 1)) != 0) break;
        v2f cc[4];
#pragma unroll
        for (int j = 0; j < 4; ++j)
          cc[j] = pk_mul(stk[depth - 2][j], stk[depth - 1][j]);
        depth -= 2;
        v2f sm = vmax2(pk_add(pk_add(cc[0], cc[1]), pk_add(cc[2], cc[3])),
                       EPS_F);
        ls = pk_add(ls, vlog2c(sm));
        v2f r = vrcp2(sm);
#pragma unroll
        for (int j = 0; j < 4; ++j) cc[j] = pk_mul(cc[j], r);
        // per-tree dot <Lk,pi>, component-wise fma chain (no shuffles)
        v2f dot = pk_mul(cc[0], pi[0]);
        dot = pk_fma(cc[1], pi[1], dot);
        dot = pk_fma(cc[2], pi[2], dot);
        dot = pk_fma(cc[3], pi[3], dot);
        if (d == 6) {
          ll = pk_add(vlog2c(vmax2(dot, EPS_F)), ls);
        } else {
          float an = Ash[LOFF[d] + (s >> d)];
          v2f an2  = splat2(an);
          v2f od   = pk_mul(splat2(1.0f - an), dot);
#pragma unroll
          for (int j = 0; j < 4; ++j)
            stk[depth][j] = pk_fma(an2, cc[j], od);
          ++depth;
        }
      }
    }
  }

  // masked diff + deterministic reduction (ll.x = pred, ll.y = true)
  float mval = vmask[site] ? 1.0f : 0.0f;
  float v = (ll.y - ll.x) * mval;
#pragma unroll
  for (int off = 16; off > 0; off >>= 1) {
    v    += __shfl_xor(v, off, 32);
    mval += __shfl_xor(mval, off, 32);
  }
  if (lane == 0) { wred[wv] = v; wred[WAVES_PB + wv] = mval; }
  __syncthreads();
  if (tid == 0) {
    float ts = 0.f, tc = 0.f;
#pragma unroll
    for (int i = 0; i < WAVES_PB; ++i) { ts += wred[i]; tc += wred[WAVES_PB + i]; }
    ws[WS_SUM + blockIdx.x] = ts;
    ws[WS_CNT + blockIdx.x] = tc;
  }
}

__global__ void f81_init_kernel(const float* __restrict__ bl,
                                float* __restrict__ ws) {
  int t = threadIdx.x;
  if (t < 2 * N_LEAVES - 2) ws[WS_A + t] = __expf(-bl[t]);   // MU = 1.0
}

__global__ void f81_final_kernel(const float* __restrict__ ws,
                                 float* __restrict__ out) {
  __shared__ float ss[256], sc[256];
  int t = threadIdx.x;
  float s = 0.f, c = 0.f;
  for (int b = t; b < N_BLOCKS; b += 256) {
    s += ws[WS_SUM + b];
    c += ws[WS_CNT + b];
  }
  ss[t] = s; sc[t] = c;
  __syncthreads();
  for (int off = 128; off > 0; off >>= 1) {
    if (t < off) { ss[t] += ss[t + off]; sc[t] += sc[t + off]; }
    __syncthreads();
  }
  if (t == 0) out[0] = (sc[0] > 0.f) ? ss[0] / sc[0] : 0.0f;
}

extern "C" void kernel_launch(void* const* d_in, const int* in_sizes, int n_in,
                              void* d_out, int out_size, void* d_ws, size_t ws_size,
                              hipStream_t stream) {
  (void)in_sizes; (void)n_in; (void)out_size; (void)ws_size;
  const float*         logits = (const float*)d_in[0];          // (16,8192,4) f32
  const int*           codes  = (const int*)d_in[1];            // (16,8192,64) i32
  const float*         pitg   = (const float*)d_in[2];          // (16,8192,4) f32
  const unsigned char* mask   = (const unsigned char*)d_in[3];  // (16,8192) bool
  const float*         bl     = (const float*)d_in[4];          // (126,) f32
  float* ws = (float*)d_ws;

  f81_init_kernel<<<1, 128, 0, stream>>>(bl, ws);
  f81_main_kernel<<<N_BLOCKS, BLOCK_T, 0, stream>>>(logits, codes, pitg, mask, ws);
  f81_final_kernel<<<1, 256, 0, stream>>>(ws, (float*)d_out);
}